// GAT_24919400251710
// MI455X (gfx1250) — compile-verified
//
#include <hip/hip_runtime.h>
#include <math.h>

#define LRELU_SLOPE 0.2f

typedef float v2f __attribute__((ext_vector_type(2)));
typedef float v8f __attribute__((ext_vector_type(8)));

// ---------------- ordered-uint encoding for float atomicMax ----------------
__device__ __forceinline__ unsigned f2ord(float f) {
  unsigned u = __float_as_uint(f);
  return (u & 0x80000000u) ? ~u : (u | 0x80000000u);
}
__device__ __forceinline__ float ord2f(unsigned u) {
  return (u & 0x80000000u) ? __uint_as_float(u & 0x7FFFFFFFu) : __uint_as_float(~u);
}

// ---------------- GEMM: H[M,Fo] = X[M,K] @ W[K,Fo] via V_WMMA_F32_16X16X4_F32 ----
// One wave computes one 16x16 output tile. 4 waves per block (blockDim=128).
// grid = (M/16, ceil(Fo/64)).  M (=N nodes=20000) is a multiple of 16.
__global__ void gat_gemm_wmma(const float* __restrict__ X, const float* __restrict__ W,
                              float* __restrict__ H, int M, int K, int Fo) {
  const int lane = threadIdx.x & 31;
  const int wave = threadIdx.x >> 5;
  const int m0   = blockIdx.x << 4;
  const int col0 = ((blockIdx.y << 2) + wave) << 4;
  if (col0 >= Fo) return;                 // wave-uniform: EXEC stays all-ones for WMMA
  const int half = lane >> 4;             // 0 -> K{0,1}, 1 -> K{2,3}
  const int r    = lane & 15;
  const size_t rowBase = (size_t)(m0 + r) * (size_t)K;
  v8f c = {};
  for (int k = 0; k < K; k += 4) {
    const int ka = k + (half << 1);
    v2f a, b;
    a.x = X[rowBase + ka];
    a.y = X[rowBase + ka + 1];
    b.x = W[(size_t)ka * Fo + col0 + r];
    b.y = W[(size_t)(ka + 1) * Fo + col0 + r];
    // (neg_a, A, neg_b, B, c_mod, C, reuse_a, reuse_b)
    c = __builtin_amdgcn_wmma_f32_16x16x4_f32(false, a, false, b, (short)0, c, false, false);
  }
#pragma unroll
  for (int v = 0; v < 8; ++v)
    H[(size_t)(m0 + v + (half << 3)) * Fo + col0 + r] = c[v];
}

// ---------------- per-(node,head) attention logits: one wave per dot pair ----
__global__ void gat_alpha(const float* __restrict__ H, const float* __restrict__ aS,
                          const float* __restrict__ aD, float* __restrict__ alphaS,
                          float* __restrict__ alphaD, int Nn, int heads, int C) {
  const int gw   = (int)((blockIdx.x * (size_t)blockDim.x + threadIdx.x) >> 5);
  const int lane = threadIdx.x & 31;
  if (gw >= Nn * heads) return;
  const int node = gw / heads;
  const int head = gw - node * heads;
  const float* hrow = H + (size_t)node * heads * C + (size_t)head * C;
  float ss = 0.f, sd = 0.f;
  for (int c = lane; c < C; c += 32) {
    const float hv = hrow[c];
    ss += hv * aS[head * C + c];
    sd += hv * aD[head * C + c];
  }
#pragma unroll
  for (int off = 16; off > 0; off >>= 1) {
    ss += __shfl_down(ss, off, 32);
    sd += __shfl_down(sd, off, 32);
  }
  if (lane == 0) { alphaS[gw] = ss; alphaD[gw] = sd; }
}

// ---------------- fill 32-bit pattern ----------------
__global__ void gat_fill_u32(unsigned* __restrict__ p, unsigned val, size_t n) {
  size_t t = blockIdx.x * (size_t)blockDim.x + threadIdx.x;
  if (t < n) p[t] = val;
}

// ---------------- edge pass 1: segment max of leakyrelu(aS[src]+aD[dst]) ----
__global__ void gat_edge_max(const int* __restrict__ src, const int* __restrict__ dst,
                             const float* __restrict__ aS, const float* __restrict__ aD,
                             unsigned* __restrict__ m, int Ee, int heads) {
  int t = blockIdx.x * blockDim.x + threadIdx.x;
  if (t >= Ee * heads) return;
  const int e = t / heads, h = t - e * heads;
  float v = aS[src[e] * heads + h] + aD[dst[e] * heads + h];
  v = v > 0.f ? v : LRELU_SLOPE * v;
  atomicMax(&m[dst[e] * heads + h], f2ord(v));
}

// ---------------- edge pass 2: segment sum of exp(e - m[dst]) ----
__global__ void gat_edge_sum(const int* __restrict__ src, const int* __restrict__ dst,
                             const float* __restrict__ aS, const float* __restrict__ aD,
                             const unsigned* __restrict__ m, float* __restrict__ ssum,
                             int Ee, int heads) {
  int t = blockIdx.x * blockDim.x + threadIdx.x;
  if (t >= Ee * heads) return;
  const int e = t / heads, h = t - e * heads;
  const int d = dst[e];
  float v = aS[src[e] * heads + h] + aD[d * heads + h];
  v = v > 0.f ? v : LRELU_SLOPE * v;
  const float ex = __expf(v - ord2f(m[d * heads + h]));
  atomicAdd(&ssum[d * heads + h], ex);
}

// ---------------- edge pass 3: out[dst] += alpha * h[src]  (one wave per edge) ----
__global__ void gat_edge_aggr(const int* __restrict__ src, const int* __restrict__ dst,
                              const float* __restrict__ H, const float* __restrict__ aS,
                              const float* __restrict__ aD, const unsigned* __restrict__ m,
                              const float* __restrict__ ssum, float* __restrict__ out,
                              int Ee, int heads, int C) {
  const int gw   = (int)((blockIdx.x * (size_t)blockDim.x + threadIdx.x) >> 5);
  const int lane = threadIdx.x & 31;
  if (gw >= Ee) return;
  const int sN = src[gw], dN = dst[gw];
  float alpha[4];
  for (int h = 0; h < heads; ++h) {
    float v = aS[sN * heads + h] + aD[dN * heads + h];
    v = v > 0.f ? v : LRELU_SLOPE * v;
    const float ex = __expf(v - ord2f(m[dN * heads + h]));
    alpha[h] = ex / (ssum[dN * heads + h] + 1e-16f);
  }
  const int F = heads * C;
  for (int f = lane; f < F; f += 32) {
    atomicAdd(&out[(size_t)dN * F + f], H[(size_t)sN * F + f] * alpha[f / C]);
  }
}

// ---------------- epilogue: out = (relu?)(out + bias) ----------------
__global__ void gat_bias_act(float* __restrict__ out, const float* __restrict__ b,
                             size_t n, int F, int doRelu) {
  size_t t = blockIdx.x * (size_t)blockDim.x + threadIdx.x;
  if (t >= n) return;
  float v = out[t] + b[t % F];
  out[t] = doRelu ? fmaxf(v, 0.f) : v;
}

// ======================= host-side layer driver ==========================
static void run_gat_layer(const float* X, int K, const float* W, const float* aws,
                          const float* awd, const float* bias, int heads, int C,
                          float* H, float* alphaS, float* alphaD, unsigned* m, float* s,
                          float* out, const int* srcI, const int* dstI,
                          int Nn, int Ee, int doRelu, hipStream_t stream) {
  const int Fo = heads * C;
  // 1) dense transform (WMMA)
  dim3 gg((Nn + 15) / 16, (Fo + 63) / 64);
  gat_gemm_wmma<<<gg, 128, 0, stream>>>(X, W, H, Nn, K, Fo);
  // 2) per-node logits
  {
    size_t thr = (size_t)Nn * heads * 32;
    gat_alpha<<<(unsigned)((thr + 255) / 256), 256, 0, stream>>>(H, aws, awd, alphaS, alphaD,
                                                                 Nn, heads, C);
  }
  // 3) init m = ord(-inf) = 0x007FFFFF, s = 0, out = 0
  {
    size_t nmh = (size_t)Nn * heads;
    gat_fill_u32<<<(unsigned)((nmh + 255) / 256), 256, 0, stream>>>(m, 0x007FFFFFu, nmh);
    gat_fill_u32<<<(unsigned)((nmh + 255) / 256), 256, 0, stream>>>((unsigned*)s, 0u, nmh);
    size_t nout = (size_t)Nn * Fo;
    gat_fill_u32<<<(unsigned)((nout + 255) / 256), 256, 0, stream>>>((unsigned*)out, 0u, nout);
  }
  // 4) segment softmax (max, sum) + weighted aggregation
  {
    int teh = Ee * heads;
    gat_edge_max<<<(teh + 255) / 256, 256, 0, stream>>>(srcI, dstI, alphaS, alphaD, m, Ee, heads);
    gat_edge_sum<<<(teh + 255) / 256, 256, 0, stream>>>(srcI, dstI, alphaS, alphaD, m, s, Ee, heads);
    size_t thr = (size_t)Ee * 32;
    gat_edge_aggr<<<(unsigned)((thr + 255) / 256), 256, 0, stream>>>(srcI, dstI, H, alphaS, alphaD,
                                                                     m, s, out, Ee, heads, C);
  }
  // 5) bias (+ReLU)
  {
    size_t nout = (size_t)Nn * Fo;
    gat_bias_act<<<(unsigned)((nout + 255) / 256), 256, 0, stream>>>(out, bias, nout, Fo, doRelu);
  }
}

extern "C" void kernel_launch(void* const* d_in, const int* in_sizes, int n_in,
                              void* d_out, int out_size, void* d_ws, size_t ws_size,
                              hipStream_t stream) {
  const float* x   = (const float*)d_in[0];
  const int*   ei  = (const int*)d_in[1];
  const float* W1  = (const float*)d_in[2];
  const float* a1s = (const float*)d_in[3];
  const float* a1d = (const float*)d_in[4];
  const float* b1  = (const float*)d_in[5];
  const float* W2  = (const float*)d_in[6];
  const float* a2s = (const float*)d_in[7];
  const float* a2d = (const float*)d_in[8];
  const float* b2  = (const float*)d_in[9];
  const float* W3  = (const float*)d_in[10];
  const float* a3s = (const float*)d_in[11];
  const float* a3d = (const float*)d_in[12];
  const float* b3  = (const float*)d_in[13];

  const int HID = 64, HEADS = 4, OUTC = 64;
  const int Nn = in_sizes[0] / HID;   // 20000
  const int Ee = in_sizes[1] / 2;     // 320000
  const int* srcI = ei;
  const int* dstI = ei + Ee;

  // workspace carve-up (all float32-sized slots)
  float* h1    = (float*)d_ws;                         // [N, 256]  gemm output / act A
  float* h2    = h1 + (size_t)Nn * (HEADS * HID);      // [N, 256]  act B
  float* aS    = h2 + (size_t)Nn * (HEADS * HID);      // [N, 4]
  float* aD    = aS + (size_t)Nn * HEADS;              // [N, 4]
  unsigned* mB = (unsigned*)(aD + (size_t)Nn * HEADS); // [N, 4]
  float* sB    = (float*)(mB + (size_t)Nn * HEADS);    // [N, 4]

  // Layer 1: x[N,64] -> h2[N,256] (h1 holds the GEMM result, h2 the aggregation)
  run_gat_layer(x, HID, W1, a1s, a1d, b1, HEADS, HID,
                h1, aS, aD, mB, sB, h2, srcI, dstI, Nn, Ee, /*relu=*/1, stream);
  // Layer 2: h2[N,256] -> h2 is consumed by the GEMM into h1; aggregate back into h2?
  // h2 is still needed as GEMM input only for step 1+2 of layer 2; after that it is
  // dead, so no fresh region is needed: reusing h2 is safe
  // (edge kernels read H=h1 only). Aggregate into h2 AFTER gemm+alpha consumed it:
  run_gat_layer(h2, HEADS * HID, W2, a2s, a2d, b2, HEADS, HID,
                h1, aS, aD, mB, sB, h2, srcI, dstI, Nn, Ee, /*relu=*/1, stream);
  // Layer 3: h2[N,256] -> d_out[N,64], single head, no relu
  run_gat_layer(h2, HEADS * HID, W3, a3s, a3d, b3, 1, OUTC,
                h1, aS, aD, mB, sB, (float*)d_out, srcI, dstI, Nn, Ee, /*relu=*/0, stream);
}